// KVMemory_38371237822771
// MI455X (gfx1250) — compile-verified
//
#include <hip/hip_runtime.h>
#include <hip/hip_bf16.h>

// KV-memory attention for MI455X (gfx1250, wave32).
// One block per batch row; 4 waves; f32 WMMA (16x16x4) for the logit
// dot-products; coalesced row gathers staged via LDS; masked softmax;
// prob-weighted value gather with prefetch. Memory-(gather-)bound kernel:
// ~205MB gathered traffic, tables nearly resident in 192MB L2.

typedef __attribute__((ext_vector_type(2))) float v2f;
typedef __attribute__((ext_vector_type(8))) float v8f;

#define KVM_K 50
#define KVM_D 256
#define KE_STRIDE 260               // row pad: 16 rows x b64 fetch -> all 64 LDS banks once
#define NEG_BIG (-3.402823466e+38f) // jnp.finfo(float32).min, as in the reference

__launch_bounds__(128)
__global__ void kvmem_kernel(const int* __restrict__ keys,
                             const int* __restrict__ values,
                             const int* __restrict__ plen_arr,
                             const float* __restrict__ query,
                             const float* __restrict__ key_table,
                             const float* __restrict__ value_table,
                             float* __restrict__ out) {
  __shared__ float q_s[KVM_D];
  __shared__ float ke_s[4][16][KE_STRIDE];
  __shared__ float lg_s[64];
  __shared__ float prob_s[64];

  const int b    = blockIdx.x;
  const int t    = threadIdx.x;
  const int w    = t >> 5;   // wave id 0..3, owns keys [16w, 16w+16)
  const int lane = t & 31;
  const int plen = plen_arr[b];  // in [1, K]

  // Stage query[b] (1KB) into LDS, shared by all waves.
  q_s[t]       = query[b * KVM_D + t];
  q_s[t + 128] = query[b * KVM_D + t + 128];

  // Stage this wave's gathered key rows. Rows k >= plen have prob == 0 in the
  // reference (masked to finfo.min before softmax), so skip their gathers.
  const int kbase  = w * 16;
  const int nvalid = min(plen, KVM_K) - kbase;
  if (nvalid > 0) {
    const int nload = min(nvalid, 16);
    for (int r = 0; r < nload; ++r) {
      const int krow = keys[b * KVM_K + kbase + r];
      const float4* src = (const float4*)(key_table + (size_t)krow * KVM_D);
      float4 x0 = src[lane];        // 32 lanes x 16B = 512B coalesced
      float4 x1 = src[lane + 32];
      *(float4*)&ke_s[w][r][4 * lane]       = x0;
      *(float4*)&ke_s[w][r][128 + 4 * lane] = x1;
    }
  }
  __syncthreads();

  if (nvalid > 0) {
    // logits tile: C[16x16] = sum_j A_j[16x4] x B_j[4x16], j over D/4 chunks.
    // A layout (16x4 f32): lane L holds row M=L%16; VGPR0/1 = dims {0,1}
    // (lanes 0-15) or {2,3} (lanes 16-31) of each 4-dim chunk.
    // B = query dims broadcast across all 16 columns (same slot pattern).
    v8f acc = {};
    const int r  = lane & 15;
    const int h2 = (lane >> 4) << 1;  // 0 | 2
    const float* arow = &ke_s[w][r][h2];
    const float* qrow = &q_s[h2];
#pragma unroll 4
    for (int j = 0; j < KVM_D / 4; ++j) {
      v2f a  = *(const v2f*)(arow + 4 * j);  // ds_load_b64, bank-conflict-free
      v2f qb = *(const v2f*)(qrow + 4 * j);
      acc = __builtin_amdgcn_wmma_f32_16x16x4_f32(false, a, false, qb,
                                                  (short)0, acc, false, false);
    }
    // C layout: VGPR v = row M=v (lanes 0-15) / M=v+8 (lanes 16-31); all 16
    // columns identical (B columns were a broadcast), so read column 0/16.
    if (lane == 0) {
#pragma unroll
      for (int v = 0; v < 8; ++v) lg_s[kbase + v] = acc[v];
    }
    if (lane == 16) {
#pragma unroll
      for (int v = 0; v < 8; ++v) lg_s[kbase + 8 + v] = acc[v];
    }
  }
  __syncthreads();

  // Mask (>= plen -> finfo.min) then stable softmax over 64 slots; wave 0
  // does the reductions with wave32 shuffles.
  if (t < 64) lg_s[t] = (t < plen) ? lg_s[t] : NEG_BIG;
  __syncthreads();
  if (t < 32) {
    float a0 = lg_s[t], a1 = lg_s[t + 32];
    float m = fmaxf(a0, a1);
    for (int off = 16; off > 0; off >>= 1)
      m = fmaxf(m, __shfl_xor(m, off, 32));
    float e0 = __expf(a0 - m);
    float e1 = __expf(a1 - m);
    float s = e0 + e1;
    for (int off = 16; off > 0; off >>= 1)
      s += __shfl_xor(s, off, 32);
    float inv = 1.0f / s;
    prob_s[t]      = e0 * inv;
    prob_s[t + 32] = e1 * inv;
  }
  __syncthreads();

  // knowledge[b,:] = sum_k prob[k] * value_table[values[b,k], :]
  // 128 threads x 2 dims: each value row is a fully coalesced 1KB gather.
  const int d0 = 2 * t;
  const int vbase = b * KVM_K;
  float acc0 = 0.f, acc1 = 0.f;
  for (int k = 0; k < plen; ++k) {
    const float p   = prob_s[k];
    const int  vrow = values[vbase + k];
    const float2 vv = *(const float2*)(value_table + (size_t)vrow * KVM_D + d0);
    if (k + 1 < plen) {  // prefetch next gathered row (global_prefetch_b8)
      __builtin_prefetch(value_table + (size_t)values[vbase + k + 1] * KVM_D + d0, 0, 1);
    }
    acc0 = fmaf(p, vv.x, acc0);
    acc1 = fmaf(p, vv.y, acc1);
  }
  out[b * KVM_D + d0]     = acc0;
  out[b * KVM_D + d0 + 1] = acc1;
}

extern "C" void kernel_launch(void* const* d_in, const int* in_sizes, int n_in,
                              void* d_out, int out_size, void* d_ws, size_t ws_size,
                              hipStream_t stream) {
  // setup_inputs order: keys, values, pair_length, query, key_table, value_table
  const int*   keys        = (const int*)d_in[0];
  const int*   values      = (const int*)d_in[1];
  const int*   plen        = (const int*)d_in[2];
  const float* query       = (const float*)d_in[3];
  const float* key_table   = (const float*)d_in[4];
  const float* value_table = (const float*)d_in[5];
  float*       out         = (float*)d_out;

  const int B = in_sizes[2];  // pair_length has one entry per batch row
  kvmem_kernel<<<dim3(B), dim3(128), 0, stream>>>(keys, values, plen, query,
                                                  key_table, value_table, out);
}